// Attention_38740605010058
// MI455X (gfx1250) — compile-verified
//
#include <hip/hip_runtime.h>
#include <hip/hip_bf16.h>

#define T_DIM 4096
#define N_DIM 128
#define D_DIM 128
#define NEGV  (-1.0e9f)
#define KHALF 68    // dwords per 64-wide half-row (+4 pad): 16B-aligned, conflict-free

typedef float v2f __attribute__((ext_vector_type(2)));
typedef float v8f __attribute__((ext_vector_type(8)));
typedef int   v4i __attribute__((ext_vector_type(4)));

#if __has_builtin(__builtin_amdgcn_global_load_async_to_lds_b128) && \
    __has_builtin(__builtin_amdgcn_s_wait_asynccnt)
#define HAVE_ASYNC_LDS 1
typedef __attribute__((address_space(1))) v4i* gas_v4i;
typedef __attribute__((address_space(3))) v4i* las_v4i;
#define ASYNC_WAIT(n) do { __builtin_amdgcn_s_wait_asynccnt(n); \
                           __asm__ volatile("" ::: "memory"); } while (0)
#else
#define HAVE_ASYNC_LDS 0
#define ASYNC_WAIT(n) __asm__ volatile("" ::: "memory")
#endif

// ---------------------------------------------------------------------------
// Kernel 1: energy[n, t] = sum_d K[t,n,d] * q[n,d], masked with NEG for t>=len
// grid (16 t-chunks, 128 n), 256 threads (8 waves); each wave: 2 tiles of 16 t,
// software-pipelined in 16x64 half-tiles with 16 async b128 loads in flight.
// ---------------------------------------------------------------------------
__global__ __launch_bounds__(256) void energy_kernel(
    const float* __restrict__ q, const float* __restrict__ key,
    const int* __restrict__ lens, float* __restrict__ attn)
{
  const int n       = blockIdx.y;
  const int t_block = blockIdx.x * 256;
  const int tid     = threadIdx.x;
  const int wave    = tid >> 5;
  const int lane    = tid & 31;
  const int row     = lane & 15;          // A-matrix row (t within tile)
  const int koff    = (lane >> 4) * 2;    // K-slot 0/2 per ISA A-layout

  __shared__ float qs[D_DIM];
  __shared__ __align__(16) float ks[8][2][16 * KHALF];  // per-wave double buffer

  if (tid < D_DIM) qs[tid] = q[n * D_DIM + tid];
  __syncthreads();

  const int len = lens[n];

  // issue 8 async b128 loads for pipeline step s (tile = s>>1, half = s&1):
  // one instruction covers 2 rows x 64 floats (lane>>4 selects row parity).
  auto issue = [&](int step) {
    const int tile = step >> 1;
    const int half = step & 1;
    const int t0   = t_block + wave * 32 + tile * 16;
    const int rhi  = lane >> 4;
    const int c4   = (lane & 15) * 4;
    float* buf = ks[wave][half];
    __asm__ volatile("" ::: "memory");
#pragma unroll
    for (int i = 0; i < 8; ++i) {
      const int r = 2 * i + rhi;
      const float* gsrc = key + ((size_t)(t0 + r) * N_DIM + n) * D_DIM + half * 64 + c4;
      float*       ldst = buf + r * KHALF + c4;
#if HAVE_ASYNC_LDS
      __builtin_amdgcn_global_load_async_to_lds_b128((gas_v4i)gsrc, (las_v4i)ldst, 0, 0);
#else
      *(float4*)ldst = *(const float4*)gsrc;
#endif
    }
  };

  // 16 WMMAs over one 16x64 half-tile, accumulating into acc
  auto compute = [&](int step, v8f& acc) {
    const int half = step & 1;
    const float* buf = ks[wave][half];
#pragma unroll
    for (int dd = 0; dd < 64; dd += 4) {
      const int d0 = half * 64 + dd;
      v2f a, b;
      a.x = buf[row * KHALF + dd + koff];
      a.y = buf[row * KHALF + dd + koff + 1];
      b.x = qs[d0 + koff];
      b.y = qs[d0 + koff + 1];
      acc = __builtin_amdgcn_wmma_f32_16x16x4_f32(false, a, false, b,
                                                  (short)0, acc, false, false);
    }
  };

  // lanes 0-15 hold E[t0+m] in acc[m]; lanes 16-31 hold E[t0+8+m]
  auto emit = [&](int tile, const v8f& acc) {
    const int t0 = t_block + wave * 32 + tile * 16;
    float e = acc[0];
#pragma unroll
    for (int m = 1; m < 8; ++m)
      if ((lane & 7) == m) e = acc[m];
    if ((lane & 8) == 0) {
      const int t = t0 + (lane & 7) + ((lane >> 4) << 3);
      attn[(size_t)n * T_DIM + t] = (t < len) ? e : NEGV;
    }
  };

  issue(0);
  issue(1);

  v8f acc = {0.f, 0.f, 0.f, 0.f, 0.f, 0.f, 0.f, 0.f};
  ASYNC_WAIT(8);  compute(0, acc);  issue(2);   // step0 done, step1 in flight
  ASYNC_WAIT(8);  compute(1, acc);  issue(3);
  emit(0, acc);

  acc = (v8f){0.f, 0.f, 0.f, 0.f, 0.f, 0.f, 0.f, 0.f};
  ASYNC_WAIT(8);  compute(2, acc);
  ASYNC_WAIT(0);  compute(3, acc);
  emit(1, acc);
}

// ---------------------------------------------------------------------------
// Kernel 2: row softmax over T=4096, in place.  128 blocks x 256 threads.
// ---------------------------------------------------------------------------
__global__ __launch_bounds__(256) void softmax_kernel(float* __restrict__ attn)
{
  const int n    = blockIdx.x;
  const int tid  = threadIdx.x;
  const int wave = tid >> 5;
  const int lane = tid & 31;
  float* rowp = attn + (size_t)n * T_DIM;

  float e[16];
  float m = -3.4e38f;
#pragma unroll
  for (int i = 0; i < 16; ++i) { e[i] = rowp[tid + i * 256]; m = fmaxf(m, e[i]); }

  __shared__ float redm[8];
  __shared__ float reds[8];
#pragma unroll
  for (int off = 16; off >= 1; off >>= 1) m = fmaxf(m, __shfl_xor(m, off, 32));
  if (lane == 0) redm[wave] = m;
  __syncthreads();
  float bm = redm[0];
#pragma unroll
  for (int w = 1; w < 8; ++w) bm = fmaxf(bm, redm[w]);

  float s = 0.f;
#pragma unroll
  for (int i = 0; i < 16; ++i) { e[i] = __expf(e[i] - bm); s += e[i]; }
#pragma unroll
  for (int off = 16; off >= 1; off >>= 1) s += __shfl_xor(s, off, 32);
  if (lane == 0) reds[wave] = s;
  __syncthreads();
  float total = 0.f;
#pragma unroll
  for (int w = 0; w < 8; ++w) total += reds[w];
  const float inv = 1.0f / total;

#pragma unroll
  for (int i = 0; i < 16; ++i) rowp[tid + i * 256] = e[i] * inv;
}

// ---------------------------------------------------------------------------
// Kernel 3: partial[c, n, d] = sum over 256 t of attn[n,t]*V[t,n,d], via WMMA
// A(16x4) = replicated attention weights, B(4x16) = V tile.
// grid (16 t-chunks, 128 n), 256 threads (8 waves, 32 t each)
// ---------------------------------------------------------------------------
__global__ __launch_bounds__(256) void context_kernel(
    const float* __restrict__ value, const float* __restrict__ attn,
    float* __restrict__ partial)
{
  const int n       = blockIdx.y;
  const int t_block = blockIdx.x * 256;
  const int tid     = threadIdx.x;
  const int wave    = tid >> 5;
  const int lane    = tid & 31;
  const int col     = lane & 15;         // d within 16-wide block
  const int kbase   = (lane >> 4) * 2;   // B rows 0/2 per half-wave

  __shared__ float aw[256];
  __shared__ float red[8][D_DIM];

  aw[tid] = attn[(size_t)n * T_DIM + t_block + tid];
  __syncthreads();

  v8f acc[8];
#pragma unroll
  for (int db = 0; db < 8; ++db) acc[db] = (v8f){0.f,0.f,0.f,0.f,0.f,0.f,0.f,0.f};

#pragma unroll
  for (int g = 0; g < 8; ++g) {
    const int tl = wave * 32 + g * 4;          // local t of this 4-t group
    v2f a;
    a.x = aw[tl + kbase];
    a.y = aw[tl + kbase + 1];
    const size_t vb = (size_t)(t_block + tl + kbase) * (N_DIM * D_DIM) + (size_t)n * D_DIM;
#pragma unroll
    for (int db = 0; db < 8; ++db) {
      const int d0 = db * 16;
      v2f b;
      b.x = value[vb + d0 + col];
      b.y = value[vb + (size_t)(N_DIM * D_DIM) + d0 + col];
      acc[db] = __builtin_amdgcn_wmma_f32_16x16x4_f32(false, a, false, b,
                                                      (short)0, acc[db], false, false);
    }
  }

  // all C rows identical: lane<16 has out[d0+col] in acc[db][0]
  if (lane < 16) {
#pragma unroll
    for (int db = 0; db < 8; ++db) red[wave][db * 16 + col] = acc[db][0];
  }
  __syncthreads();

  if (tid < D_DIM) {
    float s = 0.f;
#pragma unroll
    for (int w = 0; w < 8; ++w) s += red[w][tid];
    partial[((size_t)blockIdx.x * N_DIM + n) * D_DIM + tid] = s;
  }
}

// ---------------------------------------------------------------------------
// Kernel 4: out[n,d] = sum over 16 chunks of partial[c,n,d]
// ---------------------------------------------------------------------------
__global__ __launch_bounds__(256) void reduce_kernel(
    const float* __restrict__ partial, float* __restrict__ out)
{
  const int id = blockIdx.x * 256 + threadIdx.x;   // 0..16383
  float s = 0.f;
#pragma unroll
  for (int c = 0; c < 16; ++c) s += partial[(size_t)c * (N_DIM * D_DIM) + id];
  out[id] = s;
}

extern "C" void kernel_launch(void* const* d_in, const int* in_sizes, int n_in,
                              void* d_out, int out_size, void* d_ws, size_t ws_size,
                              hipStream_t stream) {
  const float* q     = (const float*)d_in[0];
  const float* key   = (const float*)d_in[1];
  const float* value = (const float*)d_in[2];
  const int*   lens  = (const int*)d_in[3];

  float* out     = (float*)d_out;             // N*D
  float* attn    = out + N_DIM * D_DIM;       // N*T
  float* partial = (float*)d_ws;              // 16 * N * D floats = 1 MiB

  energy_kernel <<<dim3(16, 128), 256, 0, stream>>>(q, key, lens, attn);
  softmax_kernel<<<128,           256, 0, stream>>>(attn);
  context_kernel<<<dim3(16, 128), 256, 0, stream>>>(value, attn, partial);
  reduce_kernel <<<64,            256, 0, stream>>>(partial, out);
}